// MultiSimilarityLoss_83159156785954
// MI455X (gfx1250) — compile-verified
//
#include <hip/hip_runtime.h>
#include <hip/hip_bf16.h>

typedef float v2f __attribute__((ext_vector_type(2)));
typedef float v8f __attribute__((ext_vector_type(8)));

#define THRESH    0.5f
#define MARGIN    0.1f
#define SCALE_POS 0.05f
#define SCALE_NEG 0.002f
#define MS_EPS    1e-5f
#define N_ROWS    1024
#define N_DIM     512
#define N_CLS     16

// One block = one 16-anchor strip. 512 threads = 16 wave32s.
// Phase 1: sim[16 x 1024] strip via V_WMMA_F32_16X16X4_F32 into LDS.
// Phase 2: wave w reduces anchor (m0+w) over all 1024 columns, 16 classes.
__global__ __launch_bounds__(512) void ms_loss_strip(
    const float* __restrict__ feats,
    const int*   __restrict__ labels,
    float*       __restrict__ partials) {
  __shared__ float    s_sim[16 * N_ROWS];   // 64 KB
  __shared__ unsigned s_mask[N_ROWS];       // 4 KB
  __shared__ float    s_red[16];

  const int tid  = threadIdx.x;
  const int wave = tid >> 5;
  const int lane = tid & 31;
  const int m0   = blockIdx.x * 16;

  // ---- Phase 0: pack label bitmasks (bit c = labels[j][c]==1) ----
  for (int j = tid; j < N_ROWS; j += 512) {
    const int* lr = labels + j * N_CLS;
    unsigned m = 0;
#pragma unroll
    for (int c = 0; c < N_CLS; ++c) m |= (unsigned)(lr[c] == 1) << c;
    s_mask[j] = m;
  }

  // ---- Phase 1: 16x1024 sim strip. Wave w computes N-tiles {w, w+16, w+32, w+48}.
  // f32 WMMA fragment layout (ISA 7.12.2, 32-bit A 16x4 / B 4x16):
  //   lane = r + 16*h  holds elements [k + 2h, k + 2h + 1] of row r  -> one v2f load.
  const int h = lane >> 4;
  const int r = lane & 15;
  const float* arow  = feats + (size_t)(m0 + r) * N_DIM + 2 * h;
  const float* brow0 = feats + (size_t)((wave +  0) * 16 + r) * N_DIM + 2 * h;
  const float* brow1 = feats + (size_t)((wave + 16) * 16 + r) * N_DIM + 2 * h;
  const float* brow2 = feats + (size_t)((wave + 32) * 16 + r) * N_DIM + 2 * h;
  const float* brow3 = feats + (size_t)((wave + 48) * 16 + r) * N_DIM + 2 * h;

  v8f acc0 = {}, acc1 = {}, acc2 = {}, acc3 = {};
#pragma unroll 2
  for (int k = 0; k < N_DIM; k += 4) {
    v2f a  = *(const v2f*)(arow  + k);
    v2f b0 = *(const v2f*)(brow0 + k);
    v2f b1 = *(const v2f*)(brow1 + k);
    v2f b2 = *(const v2f*)(brow2 + k);
    v2f b3 = *(const v2f*)(brow3 + k);
    acc0 = __builtin_amdgcn_wmma_f32_16x16x4_f32(false, a, false, b0, (short)0, acc0, false, false);
    acc1 = __builtin_amdgcn_wmma_f32_16x16x4_f32(false, a, false, b1, (short)0, acc1, false, false);
    acc2 = __builtin_amdgcn_wmma_f32_16x16x4_f32(false, a, false, b2, (short)0, acc2, false, false);
    acc3 = __builtin_amdgcn_wmma_f32_16x16x4_f32(false, a, false, b3, (short)0, acc3, false, false);
  }
  // C/D layout: VGPR v, lane r+16h -> element (M = v + 8h, N = r)
#pragma unroll
  for (int v = 0; v < 8; ++v) {
    const int lm = v + 8 * h;
    s_sim[lm * N_ROWS + (wave +  0) * 16 + r] = acc0[v];
    s_sim[lm * N_ROWS + (wave + 16) * 16 + r] = acc1[v];
    s_sim[lm * N_ROWS + (wave + 32) * 16 + r] = acc2[v];
    s_sim[lm * N_ROWS + (wave + 48) * 16 + r] = acc3[v];
  }
  __syncthreads();

  // ---- Phase 2: wave w owns anchor i = m0 + w ----
  const float*   srow  = s_sim + wave * N_ROWS;
  const unsigned amask = s_mask[m0 + wave];
  const float NEG_INF = -__builtin_inff();
  float contrib = 0.f;

  for (int c = 0; c < N_CLS; ++c) {
    // pass 1: min_pos (where pos & sim < 1-eps), max_neg, pos count
    float minpos = __builtin_inff();
    float maxneg = NEG_INF;
    int   posany = 0;
    for (int jj = 0; jj < 32; ++jj) {
      const int   j   = jj * 32 + lane;       // conflict-free LDS pattern
      const float s   = srow[j];
      const unsigned bit = (s_mask[j] >> c) & 1u;
      if (bit) {
        if (s < 1.f - MS_EPS) { minpos = fminf(minpos, s); posany = 1; }
      } else {
        maxneg = fmaxf(maxneg, s);
      }
    }
#pragma unroll
    for (int off = 16; off >= 1; off >>= 1) {
      minpos = fminf(minpos, __shfl_xor(minpos, off, 32));
      maxneg = fmaxf(maxneg, __shfl_xor(maxneg, off, 32));
      posany |= __shfl_xor(posany, off, 32);
    }
    const float minposeff = posany ? minpos : NEG_INF;

    // pass 2: margin-filtered exp sums + selection counts
    float psum = 0.f, nsum = 0.f;
    int pany = 0, nany = 0;
    for (int jj = 0; jj < 32; ++jj) {
      const int   j   = jj * 32 + lane;
      const float s   = srow[j];
      const unsigned bit = (s_mask[j] >> c) & 1u;
      if (bit) {
        if ((s < 1.f - MS_EPS) && (s - MARGIN < maxneg)) {
          psum += expf(-SCALE_POS * (s - THRESH));
          pany = 1;
        }
      } else {
        if (s + MARGIN > minposeff) {
          nsum += expf(SCALE_NEG * (s - THRESH));
          nany = 1;
        }
      }
    }
#pragma unroll
    for (int off = 16; off >= 1; off >>= 1) {
      psum += __shfl_xor(psum, off, 32);
      nsum += __shfl_xor(nsum, off, 32);
      pany |= __shfl_xor(pany, off, 32);
      nany |= __shfl_xor(nany, off, 32);
    }
    if (((amask >> c) & 1u) && pany && nany) {
      contrib += log1pf(psum) / SCALE_POS + log1pf(nsum) / SCALE_NEG;
    }
  }
  if (lane == 0) s_red[wave] = contrib;
  __syncthreads();
  if (tid == 0) {
    float t = 0.f;
#pragma unroll
    for (int w = 0; w < 16; ++w) t += s_red[w];   // fixed order -> deterministic
    partials[blockIdx.x] = t;
  }
}

// Deterministic finalize: sum 64 block partials, count anchors, divide.
__global__ __launch_bounds__(256) void ms_finalize(
    const float* __restrict__ partials,
    const int*   __restrict__ labels,
    float*       __restrict__ out) {
  __shared__ float s_loss[256];
  __shared__ int   s_cnt[256];
  const int t = threadIdx.x;
  float l = (t < 64) ? partials[t] : 0.f;
  int   c = 0;
  for (int idx = t; idx < N_ROWS * N_CLS; idx += 256) c += (labels[idx] == 1);
  s_loss[t] = l;
  s_cnt[t]  = c;
  __syncthreads();
  for (int s = 128; s > 0; s >>= 1) {
    if (t < s) { s_loss[t] += s_loss[t + s]; s_cnt[t] += s_cnt[t + s]; }
    __syncthreads();
  }
  if (t == 0) out[0] = (s_cnt[0] > 0) ? s_loss[0] / (float)s_cnt[0] : 0.f;
}

extern "C" void kernel_launch(void* const* d_in, const int* in_sizes, int n_in,
                              void* d_out, int out_size, void* d_ws, size_t ws_size,
                              hipStream_t stream) {
  (void)in_sizes; (void)n_in; (void)out_size; (void)ws_size;
  const float* feats  = (const float*)d_in[0];
  const int*   labels = (const int*)d_in[1];
  float* out      = (float*)d_out;
  float* partials = (float*)d_ws;   // 64 floats of scratch

  ms_loss_strip<<<dim3(64), dim3(512), 0, stream>>>(feats, labels, partials);
  ms_finalize<<<dim3(1), dim3(256), 0, stream>>>(partials, labels, out);
}